// GraphEncoder_29858612642203
// MI455X (gfx1250) — compile-verified
//
#include <hip/hip_runtime.h>
#include <stdint.h>

#define N_NODES 100000
#define N_EDGES 1600000
#define HID 128
#define LDS_PITCH 132

typedef float v2f __attribute__((ext_vector_type(2)));
typedef float v8f __attribute__((ext_vector_type(8)));

// ---------------- zero workspace (accumulators) ----------------
__global__ void zero_kernel(float* __restrict__ p, long n) {
  long i = (long)blockIdx.x * blockDim.x + threadIdx.x;
  if (i < n) p[i] = 0.0f;
}

// ---------------- layer-1 scatter: agg1[dst] += x[src], deg[dst] += 1 ------
__global__ void scatter1_kernel(const float* __restrict__ x,
                                const long long* __restrict__ src,
                                const long long* __restrict__ dst,
                                float* __restrict__ agg1,
                                float* __restrict__ deg) {
  int e = blockIdx.x * blockDim.x + threadIdx.x;
  if (e >= N_EDGES) return;
  long long s = src[e], d = dst[e];
  atomicAdd(&agg1[2 * d + 0], x[2 * s + 0]);
  atomicAdd(&agg1[2 * d + 1], x[2 * s + 1]);
  atomicAdd(&deg[d], 1.0f);
}

// ---------------- layer 1: h = relu(mean_agg @ Wl1 + x @ Wr1 + b1), K=2 ----
__global__ void layer1_kernel(const float* __restrict__ x,
                              const float* __restrict__ agg1,
                              const float* __restrict__ deg,
                              const float* __restrict__ Wl1,
                              const float* __restrict__ Wr1,
                              const float* __restrict__ b1,
                              float* __restrict__ h) {
  int idx = blockIdx.x * blockDim.x + threadIdx.x;
  if (idx >= N_NODES * HID) return;
  int n = idx >> 7, j = idx & 127;
  float r  = 1.0f / fmaxf(deg[n], 1.0f);
  float a0 = agg1[2 * n + 0] * r, a1 = agg1[2 * n + 1] * r;
  float x0 = x[2 * n + 0],        x1 = x[2 * n + 1];
  float v = b1[j];
  v = fmaf(a0, Wl1[j],       v);
  v = fmaf(a1, Wl1[HID + j], v);
  v = fmaf(x0, Wr1[j],       v);
  v = fmaf(x1, Wr1[HID + j], v);
  h[idx] = fmaxf(v, 0.0f);
}

// ---------------- layer-2 scatter: agg2[dst][:] += h[src][:] ---------------
// one wave (32 lanes) per edge, 4 channels per lane (float4 gather, L2-resident)
__global__ void scatter2_kernel(const float* __restrict__ h,
                                const long long* __restrict__ src,
                                const long long* __restrict__ dst,
                                float* __restrict__ agg2) {
  int t = threadIdx.x;
  int e = blockIdx.x * (blockDim.x >> 5) + (t >> 5);
  if (e >= N_EDGES) return;
  int lane = t & 31;
  long long s = src[e], d = dst[e];
  const float4 hv = *(const float4*)(h + (long)s * HID + lane * 4);
  float* p = agg2 + (long)d * HID + lane * 4;
  atomicAdd(p + 0, hv.x);
  atomicAdd(p + 1, hv.y);
  atomicAdd(p + 2, hv.z);
  atomicAdd(p + 3, hv.w);
}

// ---------------- layer-2 GEMMs via V_WMMA_F32_16X16X4_F32 -----------------
// out[16 rows x 128 cols] per block; 8 waves, one 16-col tile each.
// acc = (agg2*rdeg) @ Wl2 + h @ Wr2, then + b2.
__global__ void gemm2_kernel(const float* __restrict__ h,
                             const float* __restrict__ agg2,
                             const float* __restrict__ deg,
                             const float* __restrict__ Wl2,
                             const float* __restrict__ Wr2,
                             const float* __restrict__ b2,
                             float* __restrict__ out) {
  __shared__ float lds_a[16 * LDS_PITCH];  // mean-aggregated tile
  __shared__ float lds_h[16 * LDS_PITCH];  // root-feature tile

  const int tid = threadIdx.x;
  const int m0  = blockIdx.x * 16;  // 100000 % 16 == 0 -> no ragged tail

  // cooperative stage of the two shared 16x128 A-tiles, folding 1/deg
  for (int i = tid; i < 16 * HID; i += 256) {
    int row = i >> 7, col = i & 127;
    long g  = (long)(m0 + row) * HID + col;
    float r = 1.0f / fmaxf(deg[m0 + row], 1.0f);
    lds_a[row * LDS_PITCH + col] = agg2[g] * r;
    lds_h[row * LDS_PITCH + col] = h[g];
  }
  __syncthreads();

  const int wave = tid >> 5;       // 0..7 -> output column tile
  const int lane = tid & 31;
  const int half = lane >> 4;      // 0: K lo pair, 1: K hi pair
  const int l16  = lane & 15;      // row (A) / col (B,C,D) within tile
  const int ncol = wave * 16 + l16;

  const float* arow = lds_a + l16 * LDS_PITCH;
  const float* hrow = lds_h + l16 * LDS_PITCH;

  v8f acc = {};
  // GEMM 1: mean-agg @ Wl2  (K = 128, step 4)
  #pragma unroll 4
  for (int k0 = 0; k0 < HID; k0 += 4) {
    int k = k0 + 2 * half;
    v2f a; a.x = arow[k];                  a.y = arow[k + 1];
    v2f b; b.x = Wl2[(long)k * HID + ncol]; b.y = Wl2[(long)(k + 1) * HID + ncol];
    acc = __builtin_amdgcn_wmma_f32_16x16x4_f32(false, a, false, b,
                                                (short)0, acc, false, false);
  }
  // GEMM 2: h @ Wr2, accumulated into the same C
  #pragma unroll 4
  for (int k0 = 0; k0 < HID; k0 += 4) {
    int k = k0 + 2 * half;
    v2f a; a.x = hrow[k];                  a.y = hrow[k + 1];
    v2f b; b.x = Wr2[(long)k * HID + ncol]; b.y = Wr2[(long)(k + 1) * HID + ncol];
    acc = __builtin_amdgcn_wmma_f32_16x16x4_f32(false, a, false, b,
                                                (short)0, acc, false, false);
  }

  // epilogue: C/D layout -> VGPR i holds M = i + 8*half, N = l16
  const float bias = b2[ncol];
  #pragma unroll
  for (int i = 0; i < 8; ++i) {
    int m = m0 + i + 8 * half;
    out[(long)m * HID + ncol] = acc[i] + bias;
  }
}

// ---------------------------------------------------------------------------
extern "C" void kernel_launch(void* const* d_in, const int* in_sizes, int n_in,
                              void* d_out, int out_size, void* d_ws, size_t ws_size,
                              hipStream_t stream) {
  const float*     x   = (const float*)d_in[0];
  const long long* ei  = (const long long*)d_in[1];  // int64 edge_index [2, E]
  const float*     Wl1 = (const float*)d_in[2];
  const float*     Wr1 = (const float*)d_in[3];
  const float*     b1  = (const float*)d_in[4];
  const float*     Wl2 = (const float*)d_in[5];
  const float*     Wr2 = (const float*)d_in[6];
  const float*     b2  = (const float*)d_in[7];
  const long long* src = ei;
  const long long* dst = ei + N_EDGES;

  float* ws   = (float*)d_ws;
  float* deg  = ws;                         // N floats
  float* agg1 = ws + N_NODES;               // 2N floats
  float* agg2 = ws + 3L * N_NODES;          // 128N floats (L2-resident, 51 MB)
  float* h    = ws + 131L * N_NODES;        // 128N floats (fully overwritten)
  float* out  = (float*)d_out;

  const long nzero = 131L * N_NODES;        // deg + agg1 + agg2
  zero_kernel<<<(int)((nzero + 255) / 256), 256, 0, stream>>>(ws, nzero);

  scatter1_kernel<<<(N_EDGES + 255) / 256, 256, 0, stream>>>(x, src, dst, agg1, deg);

  layer1_kernel<<<(N_NODES * HID + 255) / 256, 256, 0, stream>>>(
      x, agg1, deg, Wl1, Wr1, b1, h);

  scatter2_kernel<<<(N_EDGES + 7) / 8, 256, 0, stream>>>(h, src, dst, agg2);

  gemm2_kernel<<<N_NODES / 16, 256, 0, stream>>>(h, agg2, deg, Wl2, Wr2, b2, out);
}